// ChebSpectralConv3d_76132590289408
// MI455X (gfx1250) — compile-verified
//
#include <hip/hip_runtime.h>

// CDNA5 fp32 WMMA fragments
typedef __attribute__((ext_vector_type(2))) float v2f;
typedef __attribute__((ext_vector_type(8))) float v8f;

#define PI_F 3.14159265358979323846f

// ---------------------------------------------------------------------------
// Build DCT-II (Cf: 20x64) and DCT-III (Ci: 64x20) matrices, ref conventions:
//   forward: X[k] = 2*sum_n x[n] cos(pi k (2n+1)/128), then X[0]*=0.5
//   inverse: v[n] = (1/64)*sum_{k<20} X[k] cos(pi k (2n+1)/128)  (c[0]*=2 folded)
// ---------------------------------------------------------------------------
__global__ void init_dct_mats(float* __restrict__ Cf, float* __restrict__ Ci) {
    int i = blockIdx.x * blockDim.x + threadIdx.x;
    if (i < 20 * 64) {                 // Cf[k][n], row-major 20x64
        int k = i / 64, n = i % 64;
        float c = __cosf(PI_F * (float)k * (2.0f * n + 1.0f) / 128.0f);
        Cf[i] = (k == 0) ? 1.0f : 2.0f * c;
    }
    if (i < 64 * 20) {                 // Ci[n][k], row-major 64x20
        int n = i / 20, k = i % 20;
        Ci[i] = (1.0f / 64.0f) * __cosf(PI_F * (float)k * (2.0f * n + 1.0f) / 128.0f);
    }
}

// ---------------------------------------------------------------------------
// One separable-DCT stage as a WMMA f32 GEMM with transposed write-back.
//   per volume v: in  = V (P x N) row-major
//                 out = Y (MOUT x P) row-major,  Y[mo,p] = sum_n T[mo,n]*V[p,n]
// One wave (32 threads) per 16-wide p strip; MT = ceil(MOUT/16) acc tiles.
// ---------------------------------------------------------------------------
template <int N, int MOUT>
__global__ __launch_bounds__(32) void dct_stage(
    const float* __restrict__ in, float* __restrict__ out,
    const float* __restrict__ T, int P, int in_vstride, int out_vstride)
{
    constexpr int S  = (N == 64) ? 68 : 28;   // padded LDS row stride (floats):
                                              // bank-conflict free + 16B aligned
    constexpr int MT = (MOUT + 15) / 16;
    __shared__ float vt[16 * S];

    const int lane  = threadIdx.x;
    const int ptile = blockIdx.x;             // which 16-row strip of P
    const int vol   = blockIdx.y;             // (b, chan) volume index

    const float* vin = in + (long)vol * in_vstride + (long)ptile * 16 * N;

    // ---- stage 16 x N V-tile into LDS (float4, fully coalesced) ----
    constexpr int NV4 = N / 4;                // N = 64 or 20, both /4
    for (int i = lane; i < 16 * NV4; i += 32) {
        int r = i / NV4, c4 = i % NV4;
        float4 val = *(const float4*)(vin + r * N + c4 * 4);
        *(float4*)&vt[r * S + c4 * 4] = val;
    }
    __syncthreads();   // single-wave WG: cheap fence; DS in-order per wave

    // ---- K loop: A(16x4) from T, B(4x16) from LDS, MT acc tiles ----
    v8f acc[MT];
    #pragma unroll
    for (int mt = 0; mt < MT; ++mt) acc[mt] = (v8f){0,0,0,0,0,0,0,0};

    const int pl = lane & 15;                 // A row (M) / B col (p) index
    const int kh = (lane >> 4) << 1;          // lanes 0-15: K={0,1}; 16-31: K={2,3}

    #pragma unroll 4
    for (int n0 = 0; n0 < N; n0 += 4) {
        const int cc = n0 + kh;
        v2f b = *(const v2f*)&vt[pl * S + cc];        // B[k][p] = V[p][n0+k]
        #pragma unroll
        for (int mt = 0; mt < MT; ++mt) {
            int row = mt * 16 + pl;
            if (row >= MOUT) row = MOUT - 1;          // clamp: rows never stored
            v2f a = *(const v2f*)&T[row * N + cc];    // A[m][k] = T[mo][n0+k]
            acc[mt] = __builtin_amdgcn_wmma_f32_16x16x4_f32(
                false, a, false, b, (short)0, acc[mt], false, false);
        }
    }

    // ---- store D transposed into (MOUT x P) -------------------------------
    // Fully static masking: no per-element exec juggling in the epilogue.
    float* vout = out + (long)vol * out_vstride;
    const int col   = ptile * 16 + pl;
    const int rbase = (lane >> 4) * 8;        // lanes 16-31 hold M = r+8

    if constexpr (MOUT % 16 == 0) {
        // all rows valid: straight-line store clauses
        #pragma unroll
        for (int mt = 0; mt < MT; ++mt) {
            #pragma unroll
            for (int r = 0; r < 8; ++r)
                vout[(long)(mt * 16 + rbase + r) * P + col] = acc[mt][r];
        }
    } else {
        // MOUT == 20: tile 0 fully valid; tile 1 only rows 16..19,
        // which live in lanes 0-15 (rbase==0), elements r = 0..3.
        #pragma unroll
        for (int r = 0; r < 8; ++r)
            vout[(long)(rbase + r) * P + col] = acc[0][r];
        if (lane < 16) {
            #pragma unroll
            for (int r = 0; r < 4; ++r)
                vout[(long)(16 + r) * P + col] = acc[1][r];
        }
    }
}

// ---------------------------------------------------------------------------
// Per-mode channel mix: out[b,o,m] = sum_c modes[b,c,m] * w[c,o,m]
// 8000 modes, 32x32 mix, batch 2 fused (weight read exactly once).
// ---------------------------------------------------------------------------
__global__ void mode_mix(const float* __restrict__ modes,
                         const float* __restrict__ w,
                         float* __restrict__ out)
{
    constexpr int M = 20 * 20 * 20;
    int idx = blockIdx.x * blockDim.x + threadIdx.x;   // o*M + m
    if (idx >= 32 * M) return;
    int m = idx % M, o = idx / M;
    float a0 = 0.0f, a1 = 0.0f;
    #pragma unroll 4
    for (int c = 0; c < 32; ++c) {
        float wv = w[((long)(c * 32 + o)) * M + m];    // (c,o,kd,kh,kw)
        a0 = fmaf(modes[(long)(0 * 32 + c) * M + m], wv, a0);
        a1 = fmaf(modes[(long)(1 * 32 + c) * M + m], wv, a1);
    }
    out[(long)(0 * 32 + o) * M + m] = a0;
    out[(long)(1 * 32 + o) * M + m] = a1;
}

// ---------------------------------------------------------------------------
// Orchestration. Axis-cycling transposed writes:
//  F1 (b,c,d,h,w)->(kw,d,h)  F2 ->(kh,kw,d)  F3 ->(kd,kh,kw)=modes
//  mix modes(b,c)->(b,o)
//  I1 ->(w,kd,kh)  I2 ->(h,w,kd)  I3 ->(d,h,w) = final output order.
// ws: [Cf 1280][Ci 1280][pad][A 5.24M f][B 5.24M f]  (~42 MB)
// ---------------------------------------------------------------------------
extern "C" void kernel_launch(void* const* d_in, const int* in_sizes, int n_in,
                              void* d_out, int out_size, void* d_ws, size_t ws_size,
                              hipStream_t stream) {
    (void)in_sizes; (void)n_in; (void)out_size; (void)ws_size;
    const float* x = (const float*)d_in[0];   // (2,32,64,64,64)
    const float* w = (const float*)d_in[1];   // (32,32,20,20,20)
    float* out = (float*)d_out;               // (2,32,64,64,64)
    float* ws  = (float*)d_ws;

    float* Cf = ws;                           // 20x64
    float* Ci = ws + 1280;                    // 64x20
    float* A  = ws + 4096;                    // ping
    float* B  = A + 64 * 20 * 64 * 64;        // pong (5,242,880 floats each)

    init_dct_mats<<<dim3(10), 128, 0, stream>>>(Cf, Ci);

    // forward DCT, truncated to 20 modes per axis
    dct_stage<64, 20><<<dim3(256, 64), 32, 0, stream>>>(x, A, Cf, 4096, 262144, 81920);
    dct_stage<64, 20><<<dim3(80,  64), 32, 0, stream>>>(A, B, Cf, 1280, 81920, 25600);
    dct_stage<64, 20><<<dim3(25,  64), 32, 0, stream>>>(B, A, Cf, 400,  25600, 8000);

    // channel mix in mode space
    mode_mix<<<dim3((32 * 8000 + 255) / 256), 256, 0, stream>>>(A, w, B);

    // inverse DCT (zero-padding implicit: only 20 modes enter each contraction)
    dct_stage<20, 64><<<dim3(25,  64), 32, 0, stream>>>(B, A, Ci, 400,  8000,  25600);
    dct_stage<20, 64><<<dim3(80,  64), 32, 0, stream>>>(A, B, Ci, 1280, 25600, 81920);
    dct_stage<20, 64><<<dim3(256, 64), 32, 0, stream>>>(B, out, Ci, 4096, 81920, 262144);
}